// DTW_FULL_DTWCO_46342697124329
// MI455X (gfx1250) — compile-verified
//
#include <hip/hip_runtime.h>
#include <hip/hip_bf16.h>
#include <stddef.h>

typedef __attribute__((ext_vector_type(2))) float v2f;
typedef __attribute__((ext_vector_type(4))) float v4f;
typedef __attribute__((ext_vector_type(8))) float v8f;

#define DIMN 2048
#define NDIAG 4095
#define INF_F 1e30f

// ---------------- workspace layout (bytes) ----------------
#define OFF_COSTD 0u                     // 4095*2048 f32 = 33,546,240 B (diag-major cost)
#define OFF_DIRD  33554432u              // 4095*2048 u8  =  8,386,560 B (diag-major dirs)
#define OFF_RK    41943040u              // 2048 f32
#define OFF_RX    41951232u              // 2048 f32
#define OFF_PIS   41959424u              // 4095 i32
#define OFF_PJS   41975808u              // 4095 i32
#define OFF_PLEN  41992192u              // 1 i32
#define OFF_PSUM  41992256u              // 4095 f32

// ---------------- block reduction helper (wave32) ----------------
__device__ inline float block_reduce_sum(float v) {
    for (int off = 16; off > 0; off >>= 1)
        v += __shfl_down(v, off, 32);
    __shared__ float red[32];
    const int lane = threadIdx.x & 31;
    const int wv   = threadIdx.x >> 5;
    if (lane == 0) red[wv] = v;
    __syncthreads();
    const int nw = (blockDim.x + 31) >> 5;
    v = (threadIdx.x < nw) ? red[threadIdx.x] : 0.0f;
    if (wv == 0)
        for (int off = 16; off > 0; off >>= 1)
            v += __shfl_down(v, off, 32);
    __syncthreads();
    return v; // valid in thread 0
}

// ---------------- 1) row squared-norms for both matrices (b128 loads) ------
__global__ __launch_bounds__(256)
void rownorm_kernel(const float* __restrict__ X, const float* __restrict__ K,
                    float* __restrict__ rx, float* __restrict__ rk) {
    const float* src = (blockIdx.y == 0) ? X : K;
    float*       dst = (blockIdx.y == 0) ? rx : rk;
    const float* rowp = src + (size_t)blockIdx.x * DIMN;
    float s = 0.0f;
#pragma unroll
    for (int t = 0; t < DIMN / (256 * 4); ++t) {
        v4f v = *(const v4f*)(rowp + (t * 256 + threadIdx.x) * 4);
        s += v.x * v.x + v.y * v.y + v.z * v.z + v.w * v.w;
    }
    s = block_reduce_sum(s);
    if (threadIdx.x == 0) dst[blockIdx.x] = s;
}

// ---------------- 2) fp32 WMMA GEMM -> diagonal-major cost matrix ----------
// cost[i,j] = rk[i] + rx[j] - 2 * dot(K_i, X_j), stored at costd[(i+j)*2048 + i].
// Wave computes a 16(M) x 64(N) strip with V_WMMA_F32_16X16X4_F32.
// Fragment groups (2 K-steps = 10 v2f loads) are double-buffered so the
// next group's loads are in flight while 8 WMMAs consume the current group.
__global__ __launch_bounds__(256)
void cost_gemm_kernel(const float* __restrict__ X, const float* __restrict__ K,
                      const float* __restrict__ rx, const float* __restrict__ rk,
                      float* __restrict__ costd) {
    const int lane = threadIdx.x & 31;
    const int wave = threadIdx.x >> 5;            // 0..7
    const int m0 = blockIdx.y * 128 + wave * 16;  // kernel-row tile base (i)
    const int n0 = blockIdx.x * 64;               // input-row tile base  (j)

    const int row = lane & 15;
    const int kh  = (lane >> 4) << 1;             // 0 or 2 (K halves across lane halves)

    const float* ap  = K + (size_t)(m0 + row) * DIMN + kh;
    const float* bp0 = X + (size_t)(n0 +  0 + row) * DIMN + kh;
    const float* bp1 = X + (size_t)(n0 + 16 + row) * DIMN + kh;
    const float* bp2 = X + (size_t)(n0 + 32 + row) * DIMN + kh;
    const float* bp3 = X + (size_t)(n0 + 48 + row) * DIMN + kh;

    v8f acc0 = {}, acc1 = {}, acc2 = {}, acc3 = {};

    auto load_group = [&](v2f (&Ad)[2], v2f (&Bd)[2][4], int kk) {
#pragma unroll
        for (int u = 0; u < 2; ++u) {
            const int o = kk + 4 * u;
            Ad[u]    = *(const v2f*)(ap  + o);
            Bd[u][0] = *(const v2f*)(bp0 + o);
            Bd[u][1] = *(const v2f*)(bp1 + o);
            Bd[u][2] = *(const v2f*)(bp2 + o);
            Bd[u][3] = *(const v2f*)(bp3 + o);
        }
    };
    auto compute_group = [&](v2f (&Ad)[2], v2f (&Bd)[2][4]) {
#pragma unroll
        for (int u = 0; u < 2; ++u) {
            acc0 = __builtin_amdgcn_wmma_f32_16x16x4_f32(false, Ad[u], false, Bd[u][0],
                                                         (short)0, acc0, false, false);
            acc1 = __builtin_amdgcn_wmma_f32_16x16x4_f32(false, Ad[u], false, Bd[u][1],
                                                         (short)0, acc1, false, false);
            acc2 = __builtin_amdgcn_wmma_f32_16x16x4_f32(false, Ad[u], false, Bd[u][2],
                                                         (short)0, acc2, false, false);
            acc3 = __builtin_amdgcn_wmma_f32_16x16x4_f32(false, Ad[u], false, Bd[u][3],
                                                         (short)0, acc3, false, false);
        }
    };

    v2f Aa[2], Ba[2][4], Ab[2], Bb[2][4];
    load_group(Aa, Ba, 0);
    for (int kk = 0; kk < DIMN; kk += 16) {
        load_group(Ab, Bb, kk + 8);
        compute_group(Aa, Ba);
        if (kk + 16 < DIMN) load_group(Aa, Ba, kk + 16);
        compute_group(Ab, Bb);
    }

    // Epilogue. C/D layout: VGPR g -> M = g (lanes 0-15) or g+8 (lanes 16-31); N = lane&15.
    const int mbase = m0 + ((lane >> 4) << 3);
    const int nloc  = lane & 15;
#pragma unroll
    for (int t = 0; t < 4; ++t) {
        const v8f acc = (t == 0) ? acc0 : (t == 1) ? acc1 : (t == 2) ? acc2 : acc3;
        const int j   = n0 + t * 16 + nloc;
        const float rxv = rx[j];
#pragma unroll
        for (int g = 0; g < 8; ++g) {
            const int i = mbase + g;
            const float c = rk[i] + rxv - 2.0f * acc[g];
            costd[(size_t)(i + j) * DIMN + i] = c;  // diagonal-major
        }
    }
}

// ---------------- 3) wavefront DTW DP (single workgroup, LDS diagonals) ----
// Diagonal k+1's cost values are register-prefetched right after the barrier
// so the L2 latency overlaps the LDS work + barrier instead of sitting on the
// 4095-step serial chain.
__global__ __launch_bounds__(1024)
void dtw_dp_kernel(const float* __restrict__ costd, unsigned char* __restrict__ dird) {
    __shared__ float buf[3][DIMN];
    const int tid = threadIdx.x;
#pragma unroll
    for (int s = 0; s < 2; ++s) {
        buf[0][tid + s * 1024] = INF_F;
        buf[1][tid + s * 1024] = INF_F;
    }
    __syncthreads();

    // preload diagonal 0
    float cc[2];
#pragma unroll
    for (int s = 0; s < 2; ++s) {
        const int i = tid + s * 1024;
        cc[s] = (i == 0) ? costd[0] : 0.0f;
    }

    int r2 = 0, r1 = 1, rc = 2;  // diag k-2, k-1, current
    for (int k = 0; k < NDIAG; ++k) {
        // prefetch diagonal k+1 into registers (issued before compute/barrier)
        float cn[2] = {0.0f, 0.0f};
        if (k + 1 < NDIAG) {
            const float* nrow = costd + (size_t)(k + 1) * DIMN;
#pragma unroll
            for (int s = 0; s < 2; ++s) {
                const int i = tid + s * 1024;
                const int j = k + 1 - i;
                if (j >= 0 && j < DIMN) cn[s] = nrow[i];
            }
        }

        unsigned char* drow = dird + (size_t)k * DIMN;
#pragma unroll
        for (int s = 0; s < 2; ++s) {
            const int i = tid + s * 1024;
            const int j = k - i;
            if (j >= 0 && j < DIMN) {
                const float up   = (i > 0) ? buf[r1][i - 1] : INF_F;           // D[i-1,j]
                const float left = (j > 0) ? buf[r1][i]     : INF_F;           // D[i,j-1]
                const float dg   = (i > 0 && j > 0) ? buf[r2][i - 1] : INF_F;  // D[i-1,j-1]
                float best = fminf(fminf(dg, up), left);
                // first-min tie-break: diag, then up, then left (matches jnp.argmin)
                const int dir = (dg <= up && dg <= left) ? 0 : ((up <= left) ? 1 : 2);
                if (i == 0 && j == 0) best = 0.0f;
                buf[rc][i] = cc[s] + best;
                drow[i] = (unsigned char)dir;
            } else {
                buf[rc][i] = INF_F;
            }
        }
        __syncthreads();
        cc[0] = cn[0];
        cc[1] = cn[1];
        const int t = r2; r2 = r1; r1 = rc; rc = t;
    }
}

// ---------------- 4) sequential greedy backtrack ----------------
__global__ void backtrack_kernel(const unsigned char* __restrict__ dird,
                                 int* __restrict__ pis, int* __restrict__ pjs,
                                 int* __restrict__ plen) {
    if (threadIdx.x == 0 && blockIdx.x == 0) {
        int i = DIMN - 1, j = DIMN - 1, len = 0;
        for (;;) {
            pis[len] = i; pjs[len] = j; ++len;
            if (i == 0 && j == 0) break;
            if (i == 0)      { --j; }
            else if (j == 0) { --i; }
            else {
                const int d = dird[(size_t)(i + j) * DIMN + i];
                if (d != 2) --i;
                if (d != 1) --j;
            }
        }
        *plen = len;
    }
}

// ---------------- 5) per-path-entry squared distance (b128 loads) ----------
__global__ __launch_bounds__(256)
void pair_loss_kernel(const float* __restrict__ X, const float* __restrict__ K,
                      const int* __restrict__ pis, const int* __restrict__ pjs,
                      const int* __restrict__ plen, float* __restrict__ psum) {
    const int p   = blockIdx.x;
    const int len = *plen;
    float s = 0.0f;
    if (p < len) {
        const float* kr = K + (size_t)pis[p] * DIMN;
        const float* xr = X + (size_t)pjs[p] * DIMN;
#pragma unroll
        for (int t = 0; t < DIMN / (256 * 4); ++t) {
            const int o = (t * 256 + threadIdx.x) * 4;
            v4f kv = *(const v4f*)(kr + o);
            v4f xv = *(const v4f*)(xr + o);
            v4f d  = kv - xv;
            s += d.x * d.x + d.y * d.y + d.z * d.z + d.w * d.w;
        }
    }
    s = block_reduce_sum(s);
    if (threadIdx.x == 0) psum[p] = s;
}

// ---------------- 6) deterministic final reduction ----------------
__global__ __launch_bounds__(1024)
void final_reduce_kernel(const float* __restrict__ psum, float* __restrict__ out) {
    float s = 0.0f;
    for (int t = threadIdx.x; t < NDIAG; t += 1024) s += psum[t];
    s = block_reduce_sum(s);
    if (threadIdx.x == 0) out[0] = s;
}

extern "C" void kernel_launch(void* const* d_in, const int* in_sizes, int n_in,
                              void* d_out, int out_size, void* d_ws, size_t ws_size,
                              hipStream_t stream) {
    (void)in_sizes; (void)n_in; (void)out_size; (void)ws_size;
    const float* X = (const float*)d_in[0];  // "input"  [2048, 2048] fp32
    const float* K = (const float*)d_in[1];  // "kernel" [2048, 2048] fp32
    float* out = (float*)d_out;              // scalar fp32 loss

    char* ws = (char*)d_ws;
    float*         costd = (float*)(ws + OFF_COSTD);
    unsigned char* dird  = (unsigned char*)(ws + OFF_DIRD);
    float*         rk    = (float*)(ws + OFF_RK);
    float*         rx    = (float*)(ws + OFF_RX);
    int*           pis   = (int*)(ws + OFF_PIS);
    int*           pjs   = (int*)(ws + OFF_PJS);
    int*           plen  = (int*)(ws + OFF_PLEN);
    float*         psum  = (float*)(ws + OFF_PSUM);

    rownorm_kernel<<<dim3(DIMN, 2), 256, 0, stream>>>(X, K, rx, rk);
    cost_gemm_kernel<<<dim3(DIMN / 64, DIMN / 128), 256, 0, stream>>>(X, K, rx, rk, costd);
    dtw_dp_kernel<<<1, 1024, 0, stream>>>(costd, dird);
    backtrack_kernel<<<1, 32, 0, stream>>>(dird, pis, pjs, plen);
    pair_loss_kernel<<<NDIAG, 256, 0, stream>>>(X, K, pis, pjs, plen, psum);
    final_reduce_kernel<<<1, 1024, 0, stream>>>(psum, out);
}